// AnomalyDetector_12575664242837
// MI455X (gfx1250) — compile-verified
//
#include <hip/hip_runtime.h>
#include <stdint.h>

// GraphSAGE autoencoder for MI455X (gfx1250).
// Scatter/atomic-bound workload (edge traffic ~3 GB vs ~11.5 GFLOP of GEMM),
// GEMMs use exact-fp32 V_WMMA_F32_16X16X4_F32 with TDM (tensor_load_to_lds)
// staging of A-strips and B^T K-slices into LDS, shared by 8 waves per block.

typedef __attribute__((ext_vector_type(2))) float v2f;
typedef __attribute__((ext_vector_type(8))) float v8f;
typedef __attribute__((ext_vector_type(4))) unsigned int u32x4;
typedef __attribute__((ext_vector_type(8))) int i32x8;
typedef __attribute__((ext_vector_type(4))) int i32x4;

#define IN_CH  128
#define HID_CH 128
#define LAT_CH 64

// ---------------- TDM 2D tile load: global -> LDS ----------------
// D# per CDNA5 ISA ch.8: group0 = {flags, lds_addr, global_addr, type},
// group1 = {data_size, tensor dims/strides, tile dims}. 4-byte elements.
// tile: tile_w elems per row, tile_h rows, row pitch = stride elems.
__device__ __forceinline__ void tdm_load_2d(const float* lds_dst,
                                            const float* gsrc,
                                            uint32_t tile_w, uint32_t tile_h,
                                            uint32_t stride_elems) {
  uint64_t ga = (uint64_t)(uintptr_t)gsrc;
  uint32_t lds_off = (uint32_t)(uintptr_t)lds_dst;  // low 32 bits = LDS byte addr

  u32x4 g0;
  g0[0] = 1u;                                        // count=1, user-mode D#
  g0[1] = lds_off;                                   // lds_addr (bytes)
  g0[2] = (uint32_t)ga;                              // global_addr[31:0]
  g0[3] = (uint32_t)((ga >> 32) & 0x1FFFFFFu)        // global_addr[56:32]
        | (2u << 30);                                // type = 2 ("image")

  i32x8 g1;
  g1[0] = (int)(2u << 16);                           // data_size=2 -> 4B elems
  g1[1] = (int)((tile_w & 0xFFFFu) << 16);           // tensor_dim0[15:0] @bit48
  g1[2] = (int)((tile_w >> 16) | ((tile_h & 0xFFFFu) << 16)); // dim0 hi | dim1 lo
  g1[3] = (int)((tile_h >> 16) | ((tile_w & 0xFFFFu) << 16)); // dim1 hi | tile_dim0
  g1[4] = (int)(tile_h & 0xFFFFu);                   // tile_dim1 (tile_dim2=0)
  g1[5] = (int)stride_elems;                         // tensor_dim0_stride[31:0]
  g1[6] = 0;                                         // stride hi | dim1_stride lo
  g1[7] = 0;
  i32x4 z4 = {0, 0, 0, 0};
#if __clang_major__ >= 23
  i32x8 z8 = {0, 0, 0, 0, 0, 0, 0, 0};
  __builtin_amdgcn_tensor_load_to_lds(g0, g1, z4, z4, z8, 0);
#else
  __builtin_amdgcn_tensor_load_to_lds(g0, g1, z4, z4, 0);
#endif
}

// ---------------- small helpers ----------------

__global__ void transpose_kernel(const float* __restrict__ W, float* __restrict__ Wt,
                                 int rows, int cols) {
  int i = blockIdx.x * blockDim.x + threadIdx.x;
  if (i < rows * cols) {
    int r = i / cols, c = i % cols;
    Wt[c * rows + r] = W[r * cols + c];   // Wt is [cols=K][rows=OUT]
  }
}

__global__ void degree_kernel(const int* __restrict__ dst, float* __restrict__ deg, int E) {
  int i = blockIdx.x * blockDim.x + threadIdx.x;
  if (i < E) atomicAdd(&deg[dst[i]], 1.0f);
}

__global__ void deg_inv_kernel(float* __restrict__ deg, int n) {
  int i = blockIdx.x * blockDim.x + threadIdx.x;
  if (i < n) deg[i] = 1.0f / fmaxf(deg[i], 1.0f);
}

// One wave32 per edge; lane handles CPL consecutive channels (CH = 32*CPL).
template <int CPL>
__global__ void scatter_add_kernel(const float* __restrict__ feat,
                                   const int* __restrict__ src,
                                   const int* __restrict__ dst,
                                   float* __restrict__ acc, int nE) {
  int wave = blockIdx.x * (blockDim.x >> 5) + (threadIdx.x >> 5);
  int lane = threadIdx.x & 31;
  if (wave >= nE) return;
  const int CH = CPL * 32;
  int s = src[wave], d = dst[wave];
  const float* fp = feat + (size_t)s * CH + lane * CPL;
  float* ap = acc + (size_t)d * CH + lane * CPL;
#pragma unroll
  for (int c = 0; c < CPL; ++c) atomicAdd(ap + c, fp[c]);
}

__global__ void combine_relu_kernel(const float* __restrict__ agg,
                                    const float* __restrict__ deginv,
                                    float* __restrict__ hs,  // in: s1, out: h
                                    int CH, int n) {
  int i = blockIdx.x * blockDim.x + threadIdx.x;
  if (i < n) {
    int row = i / CH;
    float v = agg[i] * deginv[row] + hs[i];
    hs[i] = fmaxf(v, 0.0f);
  }
}

__global__ void combine_kernel(const float* __restrict__ agg,
                               const float* __restrict__ deginv,
                               const float* __restrict__ s,
                               float* __restrict__ out, int CH, int n) {
  int i = blockIdx.x * blockDim.x + threadIdx.x;
  if (i < n) {
    int row = i / CH;
    out[i] = agg[i] * deginv[row] + s[i];
  }
}

// ---------------- fp32 WMMA GEMM with TDM/LDS staging ----------------
// outA = A @ Ba^T + bias ; (DUAL) outB = A @ Bb^T.
// Bat/Bbt stored transposed [K][OUT]. Block = 256 thr = 8 waves covering
// MT M-tiles x NT N-tiles (NT = OUT/16, MT = 8/NT). Wave 0 TDM-loads the
// A strip and B^T K-slices into LDS each KC-chunk; all waves read
// fragments from LDS (ds_load) and accumulate with V_WMMA_F32_16X16X4_F32.
// Fragment layouts per ISA 7.12.2:
//   A 16x4:  lanes 0-15 -> {K=k,k+1}, lanes 16-31 -> {K=k+2,k+3}, M = lane%16
//   B 4x16:  vgpr0/1 hold rows K = k+2*half +0/+1, N = lane%16
//   C 16x16: vgpr r -> (M = r + 8*half, N = lane%16)
template <int OUT, bool DUAL>
__global__ void wmma_gemm_kernel(const float* __restrict__ A,    // [M,K]
                                 const float* __restrict__ Bat,  // [K,OUT]
                                 const float* __restrict__ Bbt,  // [K,OUT] (DUAL)
                                 const float* __restrict__ bias, // [OUT] -> outA
                                 float* __restrict__ outA,
                                 float* __restrict__ outB,
                                 int M, int K) {
  constexpr int NT = OUT / 16;   // N-tiles per block
  constexpr int MT = 8 / NT;     // M-tiles per block
  constexpr int KC = 32;         // K-chunk staged in LDS

  __shared__ float sA[MT * 16 * KC];          // A strip slice  [MT*16][KC]
  __shared__ float sBa[KC * OUT];             // Ba^T slice     [KC][OUT]
  __shared__ float sBb[DUAL ? KC * OUT : 1];  // Bb^T slice

  const int w    = threadIdx.x >> 5;
  const int lane = threadIdx.x & 31;
  const int half = lane >> 4;
  const int l16  = lane & 15;
  const int tmLocal = w / NT;
  const int n0      = (w % NT) << 4;
  const int mBlock  = blockIdx.x * (MT * 16);

  v8f ca, cb;
  float bv = bias[n0 + l16];
#pragma unroll
  for (int r = 0; r < 8; ++r) { ca[r] = bv; cb[r] = 0.0f; }

  const int aRow = tmLocal * 16 + l16;  // row within LDS A strip

  for (int k0 = 0; k0 < K; k0 += KC) {
    if (w == 0) {
      // A strip: MT*16 rows x KC cols, row pitch K
      tdm_load_2d(sA, A + (size_t)mBlock * K + k0, KC, MT * 16, K);
      // B^T slices: KC rows x OUT cols, contiguous
      tdm_load_2d(sBa, Bat + (size_t)k0 * OUT, OUT, KC, OUT);
      if (DUAL) tdm_load_2d(sBb, Bbt + (size_t)k0 * OUT, OUT, KC, OUT);
      __builtin_amdgcn_s_wait_tensorcnt(0);
    }
    __syncthreads();

#pragma unroll
    for (int kl = 0; kl < KC; kl += 4) {
      const int kk = kl + 2 * half;
      v2f a;
      a.x = sA[aRow * KC + kk];
      a.y = sA[aRow * KC + kk + 1];

      v2f ba;
      ba.x = sBa[kk * OUT + n0 + l16];
      ba.y = sBa[(kk + 1) * OUT + n0 + l16];
      ca = __builtin_amdgcn_wmma_f32_16x16x4_f32(false, a, false, ba,
                                                 (short)0, ca, false, false);
      if (DUAL) {
        v2f bb;
        bb.x = sBb[kk * OUT + n0 + l16];
        bb.y = sBb[(kk + 1) * OUT + n0 + l16];
        cb = __builtin_amdgcn_wmma_f32_16x16x4_f32(false, a, false, bb,
                                                   (short)0, cb, false, false);
      }
    }
    __syncthreads();  // protect LDS before next chunk's TDM overwrite
  }

  const int col = n0 + l16;
#pragma unroll
  for (int r = 0; r < 8; ++r) {
    size_t row = (size_t)(mBlock + tmLocal * 16 + r + 8 * half);
    outA[row * OUT + col] = ca[r];
    if (DUAL) outB[row * OUT + col] = cb[r];
  }
}

// ---------------- launch ----------------

extern "C" void kernel_launch(void* const* d_in, const int* in_sizes, int n_in,
                              void* d_out, int out_size, void* d_ws, size_t ws_size,
                              hipStream_t stream) {
  const float* x    = (const float*)d_in[0];
  const int*   ei   = (const int*)d_in[1];
  const float* W1l  = (const float*)d_in[2];
  const float* b1l  = (const float*)d_in[3];
  const float* W1r  = (const float*)d_in[4];
  const float* W2l  = (const float*)d_in[5];
  const float* b2l  = (const float*)d_in[6];
  const float* W2r  = (const float*)d_in[7];
  const float* Wdec = (const float*)d_in[8];
  const float* bdec = (const float*)d_in[9];

  const int Nn = in_sizes[0] / IN_CH;   // 100000 (divisible by 32)
  const int E  = in_sizes[1] / 2;       // 1600000
  const int* src = ei;
  const int* dst = ei + E;

  // workspace layout (floats)
  float* ws = (float*)d_ws;
  size_t o = 0;
  float* deg   = ws + o; o += (size_t)Nn;            // degree -> deg_inv (in place)
  float* tbuf  = ws + o; o += (size_t)Nn * HID_CH;   // x@Wl^T (l1), h@W2l^T (l2)
  float* sbuf  = ws + o; o += (size_t)Nn * HID_CH;   // x@Wr^T+b -> h (in place)
  float* aggb  = ws + o; o += (size_t)Nn * HID_CH;   // scatter accumulator
  float* s2    = ws + o; o += (size_t)Nn * LAT_CH;   // h@W2r^T+b2
  float* wt1l  = ws + o; o += HID_CH * IN_CH;
  float* wt1r  = ws + o; o += HID_CH * IN_CH;
  float* wt2l  = ws + o; o += LAT_CH * HID_CH;
  float* wt2r  = ws + o; o += LAT_CH * HID_CH;
  float* wtdec = ws + o; o += IN_CH * LAT_CH;

  float* xrec = (float*)d_out;                       // [N, IN_CH]
  float* zout = (float*)d_out + (size_t)Nn * IN_CH;  // [N, LAT_CH]

  hipMemsetAsync(deg, 0, (size_t)Nn * sizeof(float), stream);
  hipMemsetAsync(aggb, 0, (size_t)Nn * HID_CH * sizeof(float), stream);

  transpose_kernel<<<(HID_CH * IN_CH + 255) / 256, 256, 0, stream>>>(W1l, wt1l, HID_CH, IN_CH);
  transpose_kernel<<<(HID_CH * IN_CH + 255) / 256, 256, 0, stream>>>(W1r, wt1r, HID_CH, IN_CH);
  transpose_kernel<<<(LAT_CH * HID_CH + 255) / 256, 256, 0, stream>>>(W2l, wt2l, LAT_CH, HID_CH);
  transpose_kernel<<<(LAT_CH * HID_CH + 255) / 256, 256, 0, stream>>>(W2r, wt2r, LAT_CH, HID_CH);
  transpose_kernel<<<(IN_CH * LAT_CH + 255) / 256, 256, 0, stream>>>(Wdec, wtdec, IN_CH, LAT_CH);

  degree_kernel<<<(E + 255) / 256, 256, 0, stream>>>(dst, deg, E);
  deg_inv_kernel<<<(Nn + 255) / 256, 256, 0, stream>>>(deg, Nn);

  // ---- layer 1: sbuf = x@W1r^T + b1 ; tbuf = x@W1l^T (aggregate transformed) ----
  wmma_gemm_kernel<HID_CH, true><<<Nn / 16, 256, 0, stream>>>(
      x, wt1r, wt1l, b1l, sbuf, tbuf, Nn, IN_CH);
  scatter_add_kernel<4><<<(E + 7) / 8, 256, 0, stream>>>(tbuf, src, dst, aggb, E);
  {
    int n1 = Nn * HID_CH;
    combine_relu_kernel<<<(n1 + 255) / 256, 256, 0, stream>>>(aggb, deg, sbuf, HID_CH, n1);
  }

  // ---- layer 2 (aggregate in 64-ch transformed space: halves scatter traffic) ----
  hipMemsetAsync(aggb, 0, (size_t)Nn * LAT_CH * sizeof(float), stream);
  wmma_gemm_kernel<LAT_CH, true><<<Nn / 32, 256, 0, stream>>>(
      sbuf, wt2r, wt2l, b2l, s2, tbuf, Nn, HID_CH);
  scatter_add_kernel<2><<<(E + 7) / 8, 256, 0, stream>>>(tbuf, src, dst, aggb, E);
  {
    int n2 = Nn * LAT_CH;
    combine_kernel<<<(n2 + 255) / 256, 256, 0, stream>>>(aggb, deg, s2, zout, LAT_CH, n2);
  }

  // ---- decoder: x_recon = z @ Wdec^T + b_dec ----
  wmma_gemm_kernel<IN_CH, false><<<Nn / 16, 256, 0, stream>>>(
      zout, wtdec, nullptr, bdec, xrec, nullptr, Nn, LAT_CH);
}